// MDNWeights_24472723652814
// MI455X (gfx1250) — compile-verified
//
#include <hip/hip_runtime.h>
#include <hip/hip_bf16.h>
#include <math.h>

typedef __attribute__((ext_vector_type(2))) float v2f;
typedef __attribute__((ext_vector_type(8))) float v8f;

#define BN 32
#define NN 1024
#define DD 512
#define GG 32
#define OO 256
#define XPAD 516   // 512 + 4 floats padding: breaks 64-bank conflicts on row-stride reads

__global__ void zero_denom_kernel(float* __restrict__ denom) {
    int i = blockIdx.x * blockDim.x + threadIdx.x;
    if (i < BN * GG) denom[i] = 0.0f;
}

// ---------------------------------------------------------------------------
// Kernel 1: logits = X @ W^T + b ; softmax over G=32 ; store pi ; denom += col sums
// block = 128 threads (4 waves), covers 64 rows of flattened (B*N, D).
// WMMA f32 16x16x4: each wave owns a 16(row) x 32(group) logits tile.
// ---------------------------------------------------------------------------
__global__ void logits_softmax_kernel(const float* __restrict__ X,
                                      const float* __restrict__ W,
                                      const float* __restrict__ bias,
                                      float* __restrict__ pi,
                                      float* __restrict__ denom) {
    extern __shared__ float smem[];
    float* smX  = smem;                         // 64 x XPAD
    float* smW  = smem + 64 * XPAD;             // 32 x XPAD
    float* sred = smem + 64 * XPAD + 32 * XPAD; // 32

    const int tid  = threadIdx.x;
    const int lane = tid & 31;
    const int wave = tid >> 5;
    const int lo   = lane & 15;     // A: M index / B: N index / C: N index
    const int hi   = lane >> 4;     // selects K pair (A/B) and row half (C)
    const int row0 = blockIdx.x * 64;
    const int b    = row0 >> 10;    // 64 | 1024 -> block never straddles a batch

    // Stage X tile (64x512) into LDS, coalesced float4 (16B-aligned: XPAD*4 % 16 == 0)
    for (int i = 0; i < 64; ++i) {
        int j = i * 128 + tid;            // 8192 float4s
        int r = j >> 7;
        int c = (j & 127) << 2;
        float4 v = *(const float4*)(X + (size_t)(row0 + r) * DD + c);
        *(float4*)(smX + r * XPAD + c) = v;
    }
    // Stage W (32x512)
    for (int i = 0; i < 32; ++i) {
        int j = i * 128 + tid;            // 4096 float4s
        int r = j >> 7;
        int c = (j & 127) << 2;
        float4 v = *(const float4*)(W + (size_t)r * DD + c);
        *(float4*)(smW + r * XPAD + c) = v;
    }
    if (tid < GG) sred[tid] = 0.0f;
    __syncthreads();

    // WMMA main loop: K = 512 in steps of 4
    const int wr0 = wave * 16;
    const int kb  = hi * 2;
    v8f c0 = {}; v8f c1 = {};
    for (int k0 = 0; k0 < DD; k0 += 4) {
        v2f a  = *(const v2f*)(smX + (wr0 + lo) * XPAD + k0 + kb);       // A: rows
        v2f b0 = *(const v2f*)(smW + lo * XPAD + k0 + kb);               // B: g 0..15
        v2f b1 = *(const v2f*)(smW + (lo + 16) * XPAD + k0 + kb);        // B: g 16..31
        c0 = __builtin_amdgcn_wmma_f32_16x16x4_f32(false, a, false, b0, (short)0, c0, false, false);
        c1 = __builtin_amdgcn_wmma_f32_16x16x4_f32(false, a, false, b1, (short)0, c1, false, false);
    }

    // Softmax over the 32 groups of each row.
    // Row m's logits live across lanes lo=0..15 of one half-wave, in c0[r], c1[r].
    const float bias0 = bias[lo];
    const float bias1 = bias[lo + 16];
    float cs0 = 0.0f, cs1 = 0.0f;
    for (int r = 0; r < 8; ++r) {
        float l0 = c0[r] + bias0;
        float l1 = c1[r] + bias1;
        float m = fmaxf(l0, l1);
        for (int msk = 1; msk <= 8; msk <<= 1)
            m = fmaxf(m, __shfl_xor(m, msk, 32));   // stays inside 16-lane group
        float e0 = __expf(l0 - m);
        float e1 = __expf(l1 - m);
        float s = e0 + e1;
        for (int msk = 1; msk <= 8; msk <<= 1)
            s += __shfl_xor(s, msk, 32);
        float inv = 1.0f / s;
        float p0 = e0 * inv;
        float p1 = e1 * inv;
        int rowg = row0 + wr0 + r + hi * 8;
        pi[(size_t)rowg * GG + lo]      = p0;
        pi[(size_t)rowg * GG + lo + 16] = p1;
        cs0 += p0;
        cs1 += p1;
    }
    // fold the two 8-row halves, then cross-wave via LDS atomics
    cs0 += __shfl_xor(cs0, 16, 32);
    cs1 += __shfl_xor(cs1, 16, 32);
    if (lane < 16) {
        atomicAdd(&sred[lo],      cs0);
        atomicAdd(&sred[lo + 16], cs1);
    }
    __syncthreads();
    if (tid < GG) atomicAdd(&denom[b * GG + tid], sred[tid]);
}

__global__ void weights_kernel(const float* __restrict__ denom, float* __restrict__ out_w) {
    int i = blockIdx.x * blockDim.x + threadIdx.x;
    if (i < BN * GG) out_w[i] = denom[i] * (1.0f / (float)NN);
}

// ---------------------------------------------------------------------------
// Kernel 3: per batch:  M1 = pi^T @ mu   (32 x 256, K=1024)
//                       M2 = pi^T @ (mu*mu + sig*sig)
// then locs = M1/denom ; scales = sqrt(max(M2/denom - locs^2, 0))
// grid = (o-tiles=16, batches=32); block = 128 threads, wave -> (gtile, moment)
// ---------------------------------------------------------------------------
__global__ void moments_kernel(const float* __restrict__ X,
                               const float* __restrict__ pi,
                               const float* __restrict__ denom,
                               float* __restrict__ out) {
    __shared__ float sm[2][GG][16];     // [moment][g][o_local]
    const int tid    = threadIdx.x;
    const int lane   = tid & 31;
    const int wave   = tid >> 5;        // 0..3
    const int lo     = lane & 15;
    const int hi     = lane >> 4;
    const int kb     = hi * 2;
    const int gtile  = wave & 1;
    const int moment = wave >> 1;
    const int ob     = blockIdx.x * 16; // o tile base
    const int b      = blockIdx.y;

    const float* Xb  = X  + (size_t)b * NN * DD;
    const float* piB = pi + (size_t)b * NN * GG;
    const int g = gtile * 16 + lo;      // A-matrix M index = group

    v8f c = {};
    for (int n0 = 0; n0 < NN; n0 += 4) {
        v2f a, bv;
        a.x = piB[(size_t)(n0 + kb)     * GG + g];   // pi^T[g][n]
        a.y = piB[(size_t)(n0 + kb + 1) * GG + g];
        if (moment == 0) {                            // wave-uniform branch (EXEC stays full)
            bv.x = Xb[(size_t)(n0 + kb)     * DD + ob + lo];
            bv.y = Xb[(size_t)(n0 + kb + 1) * DD + ob + lo];
        } else {
            float mu0 = Xb[(size_t)(n0 + kb)     * DD + ob + lo];
            float sg0 = Xb[(size_t)(n0 + kb)     * DD + OO + ob + lo];
            float mu1 = Xb[(size_t)(n0 + kb + 1) * DD + ob + lo];
            float sg1 = Xb[(size_t)(n0 + kb + 1) * DD + OO + ob + lo];
            bv.x = mu0 * mu0 + sg0 * sg0;
            bv.y = mu1 * mu1 + sg1 * sg1;
        }
        c = __builtin_amdgcn_wmma_f32_16x16x4_f32(false, a, false, bv, (short)0, c, false, false);
    }

    // exchange tiles: each wave writes its 16x16 result
    for (int r = 0; r < 8; ++r)
        sm[moment][gtile * 16 + r + hi * 8][lo] = c[r];
    __syncthreads();

    // epilogue: 32 g x 16 o outputs per block
    for (int i = tid; i < GG * 16; i += 128) {
        int gg = i >> 4;
        int ol = i & 15;
        float invd = 1.0f / denom[b * GG + gg];
        float m1 = sm[0][gg][ol];
        float m2 = sm[1][gg][ol];
        float loc = m1 * invd;
        float var = m2 * invd - loc * loc;
        float sc  = sqrtf(fmaxf(var, 0.0f));
        size_t base = (size_t)(b * GG + gg) * OO + ob + ol;
        out[(size_t)BN * GG + base]                          = sc;   // scales
        out[(size_t)BN * GG + (size_t)BN * GG * OO + base]   = loc;  // locs
    }
}

extern "C" void kernel_launch(void* const* d_in, const int* in_sizes, int n_in,
                              void* d_out, int out_size, void* d_ws, size_t ws_size,
                              hipStream_t stream) {
    const float* X    = (const float*)d_in[0];   // (B, N, D) f32
    const float* W    = (const float*)d_in[1];   // (G, D)    f32
    const float* bias = (const float*)d_in[2];   // (G,)      f32
    float* out = (float*)d_out;                  // weights | scales | locs

    float* pi    = (float*)d_ws;                       // B*N*G f32 = 4 MB (L2-resident)
    float* denom = pi + (size_t)BN * NN * GG;          // B*G f32

    zero_denom_kernel<<<4, 256, 0, stream>>>(denom);

    size_t smem1 = (size_t)(64 * XPAD + 32 * XPAD + 32) * sizeof(float); // ~194 KB of 320 KB WGP LDS
    logits_softmax_kernel<<<(BN * NN) / 64, 128, smem1, stream>>>(X, W, bias, pi, denom);

    weights_kernel<<<4, 256, 0, stream>>>(denom, out);

    dim3 g3(OO / 16, BN);
    moments_kernel<<<g3, 128, 0, stream>>>(X, pi, denom, out);
}